// QLayerMultiQubit_45878840655943
// MI455X (gfx1250) — compile-verified
//
#include <hip/hip_runtime.h>
#include <hip/hip_bf16.h>

typedef float v2f __attribute__((ext_vector_type(2)));
typedef float v4f __attribute__((ext_vector_type(4)));
typedef float v8f __attribute__((ext_vector_type(8)));

#define B_TOT   65536
#define C_CH    64
#define D_DIM   256
#define NCOL    192          // C*3 output columns of the GEMM
#define XPAD    260          // padded LDS row stride for x tile (bank-conflict free b64 reads)
#define PHIPAD  196          // padded LDS row stride for phi tile
#define NTHREADS 384         // 12 waves: one 16-wide N-tile each (12*16 = 192)

__device__ __forceinline__ v2f cmul(v2f a, v2f b) {        // a * b
    return v2f{a.x * b.x - a.y * b.y, a.x * b.y + a.y * b.x};
}
__device__ __forceinline__ v2f cmulc(v2f a, v2f b) {       // a * conj(b)
    return v2f{a.x * b.x + a.y * b.y, a.y * b.x - a.x * b.y};
}
__device__ __forceinline__ v2f cadd(v2f a, v2f b) { return v2f{a.x + b.x, a.y + b.y}; }

__global__ __launch_bounds__(NTHREADS)
void qlayer_fused(const float* __restrict__ rho_re,
                  const float* __restrict__ rho_im,
                  const float* __restrict__ x,
                  const float* __restrict__ w,
                  const float* __restrict__ theta,
                  float* __restrict__ out)
{
    __shared__ float lds_x[16 * XPAD];     // 16 x 256 x-tile, padded rows
    __shared__ float lds_phi[16 * PHIPAD]; // 16 x 192 phi tile, padded rows

    const int tid = threadIdx.x;
    const int b0  = blockIdx.x * 16;       // first batch row of this block's tile

    // ---- Phase 0: stage x tile (16 rows x 256 cols, fp32) into LDS ----
    for (int i = tid; i < 16 * 64; i += NTHREADS) {
        const int r  = i >> 6;             // row 0..15
        const int kq = (i & 63) << 2;      // col 0..252 step 4
        v4f v = *(const v4f*)(x + (b0 + r) * D_DIM + kq);
        float* dst = &lds_x[r * XPAD + kq];
        *(v2f*)(dst)     = v2f{v.x, v.y};  // XPAD*4 = 1040 B, 16B-aligned too
        *(v2f*)(dst + 2) = v2f{v.z, v.w};
    }
    __syncthreads();

    // ---- Phase 1: phi tile via V_WMMA_F32_16X16X4_F32 ----
    const int lane = tid & 31;
    const int wv   = tid >> 5;             // wave 0..11
    const int n0   = wv * 16;              // this wave's 16 output columns
    const int lr   = lane & 15;            // M (for A) / N (for B)
    const int lh   = lane >> 4;            // K half: 0 -> K=0,1 ; 1 -> K=2,3

    // A-frag: x[m= lr][k + 2*lh + {0,1}]  (from LDS)
    // B-frag: B[k][n] = w[n][k] -> w[n0+lr][k + 2*lh + {0,1}]  (contiguous in k)
    const float* xrow = &lds_x[lr * XPAD + lh * 2];
    const float* wrow = w + (n0 + lr) * D_DIM + lh * 2;

    v8f acc = {};
    #pragma unroll 8
    for (int k = 0; k < D_DIM; k += 4) {
        v2f a = *(const v2f*)(xrow + k);
        v2f b = *(const v2f*)(wrow + k);
        // D = A(16x4) * B(4x16) + C, fp32
        acc = __builtin_amdgcn_wmma_f32_16x16x4_f32(
                  /*neg_a=*/false, a, /*neg_b=*/false, b,
                  /*c_mod=*/(short)0, acc, /*reuse_a=*/false, /*reuse_b=*/false);
    }

    // C/D layout: VGPR v, lane l -> M = v + 8*(l/16), N = l%16
    const float th = theta[n0 + lr];
    #pragma unroll
    for (int v = 0; v < 8; ++v) {
        const int m = v + lh * 8;
        lds_phi[m * PHIPAD + n0 + lr] = acc[v] + th;
    }
    __syncthreads();

    // ---- Phase 2: U = f(phi), rho_out = U rho U^H, per (b_local, c) ----
    for (int idx = tid; idx < 16 * C_CH; idx += NTHREADS) {
        const int bl = idx >> 6;           // local batch row
        const int c  = idx & 63;           // channel
        const float* ph = &lds_phi[bl * PHIPAD + c * 3];
        const float p0 = ph[0], p1 = ph[1], p2 = ph[2];

        float s0, c0, sp, cp, sm, cm;
        __sincosf(0.5f * p0,        &s0, &c0);
        __sincosf(0.5f * (p1 + p2), &sp, &cp);
        __sincosf(0.5f * (p1 - p2), &sm, &cm);

        const v2f U00 = v2f{ c0 * cp,  c0 * sp};
        const v2f U01 = v2f{-s0 * cm, -s0 * sm};
        const v2f U10 = v2f{ s0 * cm, -s0 * sm};
        const v2f U11 = v2f{ c0 * cp, -c0 * sp};

        const int base = ((b0 + bl) * C_CH + c) * 4;   // 4 floats per 2x2 tile
        const v4f re = *(const v4f*)(rho_re + base);
        const v4f im = *(const v4f*)(rho_im + base);
        const v2f r00 = v2f{re.x, im.x}, r01 = v2f{re.y, im.y};
        const v2f r10 = v2f{re.z, im.z}, r11 = v2f{re.w, im.w};

        // T = U * rho
        const v2f T00 = cadd(cmul(U00, r00), cmul(U01, r10));
        const v2f T01 = cadd(cmul(U00, r01), cmul(U01, r11));
        const v2f T10 = cadd(cmul(U10, r00), cmul(U11, r10));
        const v2f T11 = cadd(cmul(U10, r01), cmul(U11, r11));
        // out[i][j] = sum_n T[i][n] * conj(U[j][n])
        const v2f O00 = cadd(cmulc(T00, U00), cmulc(T01, U01));
        const v2f O01 = cadd(cmulc(T00, U10), cmulc(T01, U11));
        const v2f O10 = cadd(cmulc(T10, U00), cmulc(T11, U01));
        const v2f O11 = cadd(cmulc(T10, U10), cmulc(T11, U11));

        // complex64 interleaved (re, im): 8 floats per (b,c), 32B-aligned
        float* o = out + (size_t)base * 2;
        *(v4f*)(o)     = v4f{O00.x, O00.y, O01.x, O01.y};
        *(v4f*)(o + 4) = v4f{O10.x, O10.y, O11.x, O11.y};
    }
}

extern "C" void kernel_launch(void* const* d_in, const int* in_sizes, int n_in,
                              void* d_out, int out_size, void* d_ws, size_t ws_size,
                              hipStream_t stream) {
    const float* rho_re = (const float*)d_in[0];
    const float* rho_im = (const float*)d_in[1];
    const float* x      = (const float*)d_in[2];
    const float* w      = (const float*)d_in[3];
    const float* theta  = (const float*)d_in[4];
    float* out = (float*)d_out;

    dim3 grid(B_TOT / 16);
    dim3 block(NTHREADS);
    hipLaunchKernelGGL(qlayer_fused, grid, block, 0, stream,
                       rho_re, rho_im, x, w, theta, out);
}